// LightGCN_67095979098843
// MI455X (gfx1250) — compile-verified
//
#include <hip/hip_runtime.h>
#include <hip/hip_bf16.h>

#define BLK 256
#define WPB 8      // waves per block (wave32)
#define D   32     // embedding dim == wave width

#define AS1 __attribute__((address_space(1)))
#define AS3 __attribute__((address_space(3)))

// ---------------- CDNA5 async global->LDS helpers ----------------
__device__ __forceinline__ void async_g2l_b32(const void* g, void* l) {
#if __has_builtin(__builtin_amdgcn_global_load_async_to_lds_b32)
    __builtin_amdgcn_global_load_async_to_lds_b32(
        (AS1 int*)(void*)g, (AS3 int*)l, 0, 0);
#else
    unsigned loff = (unsigned)(unsigned long long)(AS3 char*)l;
    asm volatile("global_load_async_to_lds_b32 %0, %1, off"
                 :: "v"(loff), "v"(g) : "memory");
#endif
}

template <int N>
__device__ __forceinline__ void wait_asynccnt() {
#if __has_builtin(__builtin_amdgcn_s_wait_asynccnt)
    __builtin_amdgcn_s_wait_asynccnt(N);
#else
    asm volatile("s_wait_asynccnt %0" :: "i"(N) : "memory");
#endif
    asm volatile("" ::: "memory");   // keep LDS reads after the wait
}

// ---------------- main propagation (non-template, defined FIRST so the
// disasm snippet shows the async-to-LDS path) ----------------
// lane = feature (d=32 == wave32); wave = edge. Edge stream is double-buffered
// through LDS with async global->LDS copies (ASYNCcnt). Each edge slot is an
// int4 {src, dst, norm_bits, pad} so the consume side is one uniform
// ds_load_b128 per edge. Gather is a coalesced 128B load (L2-resident table),
// scatter is a coalesced burst of global_atomic_add_f32 (no return).
__global__ __launch_bounds__(BLK) void k_prop_norm(const int*  __restrict__ src,
                                                   const int*  __restrict__ dst,
                                                   const float* __restrict__ nrm,
                                                   const float* __restrict__ xin,
                                                   float*       __restrict__ xout,
                                                   int E) {
    __shared__ int4 tile[2][WPB][32];

    const int lane = threadIdx.x & 31;
    const int wv   = threadIdx.x >> 5;
    const long long stride = (long long)gridDim.x * WPB * 32;
    long long base = ((long long)blockIdx.x * WPB + wv) * 32;
    if (base >= E) return;

    // issue one 32-edge stage into buffer bi (indices clamped so EXEC stays
    // full and the ASYNCcnt bookkeeping is exact)
    auto issue = [&](long long b0, int bi) {
        long long idx = b0 + lane;
        if (idx >= E) idx = (long long)E - 1;
        int4* slot = &tile[bi][wv][lane];
        async_g2l_b32(&src[idx], &slot->x);
        async_g2l_b32(&dst[idx], &slot->y);
        async_g2l_b32(&nrm[idx], &slot->z);
    };

    int buf = 0;
    issue(base, 0);

    for (; base < E; base += stride) {
        long long nb = base + stride;
        bool pref = nb < E;
        if (pref) issue(nb, buf ^ 1);

        // current stage's copies complete when ASYNCcnt drops to the
        // prefetch's outstanding count (in-order completion).
        if (pref) wait_asynccnt<3>();
        else      wait_asynccnt<0>();

        long long rem = (long long)E - base;
        int cnt = rem < 32 ? (int)rem : 32;

#pragma unroll 8
        for (int k = 0; k < cnt; ++k) {
            int4 e = tile[buf][wv][k];        // uniform ds_load_b128
            float w = __int_as_float(e.z);
            float v = xin[e.x * D + lane] * w;          // coalesced 128B gather
            atomicAdd(&xout[e.y * D + lane], v);        // global_atomic_add_f32
        }
        buf ^= 1;
    }
}

// fallback used only if ws is too small to cache per-edge norms: recompute
// rsqrt(deg[s]*deg[d]) on the fly (deg is L2-resident, uniform per edge)
__global__ __launch_bounds__(BLK) void k_prop_deg(const int*  __restrict__ src,
                                                  const int*  __restrict__ dst,
                                                  const float* __restrict__ deg,
                                                  const float* __restrict__ xin,
                                                  float*       __restrict__ xout,
                                                  int E) {
    __shared__ int4 tile[2][WPB][32];

    const int lane = threadIdx.x & 31;
    const int wv   = threadIdx.x >> 5;
    const long long stride = (long long)gridDim.x * WPB * 32;
    long long base = ((long long)blockIdx.x * WPB + wv) * 32;
    if (base >= E) return;

    auto issue = [&](long long b0, int bi) {
        long long idx = b0 + lane;
        if (idx >= E) idx = (long long)E - 1;
        int4* slot = &tile[bi][wv][lane];
        async_g2l_b32(&src[idx], &slot->x);
        async_g2l_b32(&dst[idx], &slot->y);
    };

    int buf = 0;
    issue(base, 0);

    for (; base < E; base += stride) {
        long long nb = base + stride;
        bool pref = nb < E;
        if (pref) issue(nb, buf ^ 1);
        if (pref) wait_asynccnt<2>();
        else      wait_asynccnt<0>();

        long long rem = (long long)E - base;
        int cnt = rem < 32 ? (int)rem : 32;

#pragma unroll 8
        for (int k = 0; k < cnt; ++k) {
            int4 e = tile[buf][wv][k];
            float dp = deg[e.x] * deg[e.y];
            float w = dp > 0.0f ? rsqrtf(dp) : 0.0f;
            float v = xin[e.x * D + lane] * w;
            atomicAdd(&xout[e.y * D + lane], v);
        }
        buf ^= 1;
    }
}

// ---------------- small elementwise kernels ----------------
__global__ __launch_bounds__(BLK) void k_zero_f(float* p, long long n) {
    long long i = (long long)blockIdx.x * BLK + threadIdx.x;
    long long s = (long long)gridDim.x * BLK;
    for (; i < n; i += s) p[i] = 0.0f;
}

__global__ __launch_bounds__(BLK) void k_zero_v4(float4* p, long long n4) {
    long long i = (long long)blockIdx.x * BLK + threadIdx.x;
    long long s = (long long)gridDim.x * BLK;
    float4 z = make_float4(0.f, 0.f, 0.f, 0.f);
    for (; i < n4; i += s) p[i] = z;
}

// x_cur = concat(user_emb, item_emb); acc(=d_out) = same
__global__ __launch_bounds__(BLK) void k_init(const float4* __restrict__ u,
                                              const float4* __restrict__ it,
                                              float4* __restrict__ xcur,
                                              float4* __restrict__ acc,
                                              long long nu4, long long n4) {
    long long i = (long long)blockIdx.x * BLK + threadIdx.x;
    long long s = (long long)gridDim.x * BLK;
    for (; i < n4; i += s) {
        float4 v = (i < nu4) ? u[i] : it[i - nu4];
        xcur[i] = v;
        acc[i]  = v;
    }
}

// acc = (acc + x) * mul   (mul = 1 for layers 0..1, 0.25 fused on the last)
__global__ __launch_bounds__(BLK) void k_add_scale(float4* __restrict__ acc,
                                                   const float4* __restrict__ x,
                                                   float mul, long long n4) {
    long long i = (long long)blockIdx.x * BLK + threadIdx.x;
    long long s = (long long)gridDim.x * BLK;
    for (; i < n4; i += s) {
        float4 a = acc[i], b = x[i];
        a.x = (a.x + b.x) * mul; a.y = (a.y + b.y) * mul;
        a.z = (a.z + b.z) * mul; a.w = (a.w + b.w) * mul;
        acc[i] = a;
    }
}

// ---------------- degree + norm ----------------
__global__ __launch_bounds__(BLK) void k_deg(const int* __restrict__ dst,
                                             float* __restrict__ deg, int E) {
    int e = blockIdx.x * BLK + threadIdx.x;
    if (e < E) atomicAdd(&deg[dst[e]], 1.0f);
}

__global__ __launch_bounds__(BLK) void k_norm(const int* __restrict__ src,
                                              const int* __restrict__ dst,
                                              const float* __restrict__ deg,
                                              float* __restrict__ nrm, int E) {
    int e = blockIdx.x * BLK + threadIdx.x;
    if (e < E) {
        float dp = deg[src[e]] * deg[dst[e]];
        nrm[e] = dp > 0.0f ? rsqrtf(dp) : 0.0f;
    }
}

// ---------------- host launcher ----------------
extern "C" void kernel_launch(void* const* d_in, const int* in_sizes, int n_in,
                              void* d_out, int out_size, void* d_ws, size_t ws_size,
                              hipStream_t stream) {
    (void)n_in; (void)out_size;

    const int*   edge = (const int*)d_in[0];        // [2, E] int32 (JAX x64 off)
    const float* uemb = (const float*)d_in[3];
    const float* iemb = (const float*)d_in[4];

    const int E = in_sizes[0] / 2;
    const long long nu = in_sizes[3] / D;
    const long long ni = in_sizes[4] / D;
    const long long n  = nu + ni;
    const long long nd = n * D;
    const long long nd4 = nd / 4;          // d=32 -> always divisible by 4
    const long long nu4 = nu * (D / 4);

    float* out = (float*)d_out;
    const int* srcI = edge;
    const int* dstI = edge + (size_t)E;

    // carve workspace: x_cur, x_next, deg, [norm]
    char* ws = (char*)d_ws;
    size_t off = 0;
    auto carve = [&](size_t bytes) -> void* {
        void* p = ws + off;
        off += (bytes + 255) & ~(size_t)255;
        return p;
    };
    float* xA  = (float*)carve((size_t)nd * 4);
    float* xB  = (float*)carve((size_t)nd * 4);
    float* deg = (float*)carve((size_t)n * 4);
    size_t needNorm = off + (((size_t)E * 4 + 255) & ~(size_t)255);
    bool haveNorm = ws_size >= needNorm;
    float* nrm = haveNorm ? (float*)carve((size_t)E * 4) : nullptr;

    const int ewBlocks = (E + BLK - 1) / BLK;
    int vecBlocks = (int)((nd4 + BLK - 1) / BLK);
    if (vecBlocks < 1) vecBlocks = 1;
    int nBlocks = (int)((n + BLK - 1) / BLK);
    long long tiles = ((long long)E + (WPB * 32) - 1) / (WPB * 32);
    int propBlocks = (int)(tiles < 4096 ? tiles : 4096);
    if (propBlocks < 1) propBlocks = 1;

    // 1) degrees
    k_zero_f<<<nBlocks, BLK, 0, stream>>>(deg, n);
    k_deg<<<ewBlocks, BLK, 0, stream>>>(dstI, deg, E);

    // 2) per-edge norms (cached if ws allows)
    if (haveNorm)
        k_norm<<<ewBlocks, BLK, 0, stream>>>(srcI, dstI, deg, nrm, E);

    // 3) x0 and accumulator (accumulator lives in d_out)
    k_init<<<vecBlocks, BLK, 0, stream>>>((const float4*)uemb, (const float4*)iemb,
                                          (float4*)xA, (float4*)out, nu4, nd4);

    // 4) three propagation layers, acc += layer, final fused *0.25
    float* cur = xA;
    float* nxt = xB;
    for (int l = 0; l < 3; ++l) {
        k_zero_v4<<<vecBlocks, BLK, 0, stream>>>((float4*)nxt, nd4);
        if (haveNorm)
            k_prop_norm<<<propBlocks, BLK, 0, stream>>>(srcI, dstI, nrm,
                                                        cur, nxt, E);
        else
            k_prop_deg<<<propBlocks, BLK, 0, stream>>>(srcI, dstI, deg,
                                                       cur, nxt, E);
        float mul = (l == 2) ? 0.25f : 1.0f;
        k_add_scale<<<vecBlocks, BLK, 0, stream>>>((float4*)out, (const float4*)nxt,
                                                   mul, nd4);
        float* t = cur; cur = nxt; nxt = t;
    }
}